// CatEmbedder_74165495267876
// MI455X (gfx1250) — compile-verified
//
#include <hip/hip_runtime.h>

// ---------------------------------------------------------------------------
// CatEmbedder fused kernel for MI455X (gfx1250, wave32).
//
// L2-resident embedding gathers dominate (table = 25.6MB << 192MB L2; HBM
// compulsory traffic ~40MB -> ~2us floor at 23.3TB/s).  All matmuls
// (emb@ga_W per field, 4x dense 64x64 layers) run on V_WMMA_F32_16X16X4_F32
// at full f32 precision.  One wave processes 16 batch rows end-to-end;
// ga_W B-fragments stay register-resident (reused 51x per row-group);
// summed/sumsq accumulate directly in WMMA A-fragment layout so the FM
// branch needs no transpose.  Pass 1 (compulsory-miss pass) prefetches two
// fields ahead via global_prefetch_b8, with the index stream software-
// pipelined through registers so the prefetch address never forces a
// dependent s_wait_loadcnt inside the loop.
// ---------------------------------------------------------------------------

typedef __attribute__((ext_vector_type(2))) float v2f;
typedef __attribute__((ext_vector_type(8))) float v8f;

#define B_TOTAL      32768
#define F_CAT        50
#define D_DIM        64
#define PROBE_C      39.0f
#define INV_DEN      (1.0f / (51.0f + 39.0f))
#define INV_NF       (1.0f / 51.0f)
#define WAVES_PER_WG 4
#define LDS_STRIDE   68   // floats; conflict-free b32 store / b64 load on 64 banks

static __device__ inline v8f wmma4(v2f a, v2f b, v8f c) {
  // D = A(16x4, f32) * B(4x16, f32) + C(16x16, f32)
  return __builtin_amdgcn_wmma_f32_16x16x4_f32(
      /*neg_a=*/false, a, /*neg_b=*/false, b,
      /*c_mod=*/(short)0, c, /*reuse_a=*/false, /*reuse_b=*/false);
}

// Store 16x64 result tiles (C-layout: VGPR r -> M=r+8h, lane&15 -> N) to LDS.
static __device__ inline void c_store_lds(float* xb, int m, int h, const v8f* cc) {
#pragma unroll
  for (int t = 0; t < 4; ++t)
#pragma unroll
    for (int r = 0; r < 8; ++r)
      xb[(r + 8 * h) * LDS_STRIDE + 16 * t + m] = cc[t][r];
}

// Load 16x64 matrix from LDS as WMMA A-fragments (lane m row, K pair 4s+2h).
static __device__ inline void a_load_lds(const float* xb, int m, int h, v2f* a) {
#pragma unroll
  for (int s = 0; s < 16; ++s)
    a[s] = *(const v2f*)(xb + m * LDS_STRIDE + 4 * s + 2 * h);
}

// out(16x64) = A(16x64) @ W^T + bias, optional relu.  W is DxD row-major,
// so B[k][n] = W[n][k] -> contiguous b64 loads W[n][k0], W[n][k0+1].
static __device__ inline void dense64(const v2f* a, const float* __restrict__ W,
                                      const float* __restrict__ bias,
                                      int m, int h, bool relu, v8f* out) {
#pragma unroll
  for (int t = 0; t < 4; ++t) {
    const int n = 16 * t + m;
    const float* wr = W + n * 64 + 2 * h;
    v8f c = {};
#pragma unroll
    for (int s = 0; s < 16; ++s) {
      v2f b = *(const v2f*)(wr + 4 * s);
      c = wmma4(a[s], b, c);
    }
    const float bv = bias[n];
#pragma unroll
    for (int j = 0; j < 8; ++j) {
      float v = c[j] + bv;
      out[t][j] = relu ? fmaxf(v, 0.0f) : v;
    }
  }
}

__global__ __launch_bounds__(WAVES_PER_WG * 32)
void catembed_fused(const int* __restrict__ cat_indices,     // [B,F]
                    const float* __restrict__ num_features,  // [B]
                    const float* __restrict__ embed_table,   // [NC,64]
                    const float* __restrict__ num_W,         // [64]
                    const float* __restrict__ num_b,         // [64]
                    const float* __restrict__ ga_W,          // [64,64]
                    const float* __restrict__ ga_b,          // [64]
                    const float* __restrict__ gW,            // [2,64,64]
                    const float* __restrict__ gb,            // [2,64]
                    const float* __restrict__ lW,            // [2,64,64]
                    const float* __restrict__ lb,            // [2,64]
                    float* __restrict__ out)                 // [B,64]
{
  __shared__ float xpose[WAVES_PER_WG][16 * LDS_STRIDE];

  const int wave = threadIdx.x >> 5;
  const int lane = threadIdx.x & 31;
  const int m    = lane & 15;   // row within 16-row tile / N within tile
  const int h    = lane >> 4;   // half-wave: selects K sub-pair
  const int row0 = (blockIdx.x * WAVES_PER_WG + wave) * 16;
  const int row  = row0 + m;
  float* xb = xpose[wave];

  // ---- ga_W B-fragments, resident in registers (reused for all 51 fields) ----
  // B(4x16) layout mirrors A: lane n<16 holds B[k0][n],B[k0+1][n];
  // lane n+16 holds B[k0+2][n],B[k0+3][n].  ga_W row-major [k][e] -> 2 b32.
  v2f bGA[4][16];
#pragma unroll
  for (int t = 0; t < 4; ++t)
#pragma unroll
    for (int s = 0; s < 16; ++s) {
      const int k0 = 4 * s + 2 * h;
      const int e  = 16 * t + m;
      bGA[t][s].x = ga_W[k0 * 64 + e];
      bGA[t][s].y = ga_W[(k0 + 1) * 64 + e];
    }

  // ---- numeric embedding -> init summed; sumsq starts at 0 (shallow only) ----
  const float nf = num_features[row];
  v2f summed[16], sumsq[16];
#pragma unroll
  for (int s = 0; s < 16; ++s) {
    const int d0 = 4 * s + 2 * h;
    v2f w  = *(const v2f*)(num_W + d0);
    v2f bb = *(const v2f*)(num_b + d0);
    summed[s] = nf * w + bb;
    sumsq[s]  = (v2f){0.0f, 0.0f};
  }

  const long long idxBase = (long long)row * F_CAT;

  // ---- pass 1: stream shallow fields, accumulate summed & sumsq ----
  // This pass takes the compulsory HBM misses on the table.  The index
  // stream is software-pipelined (idx0=f, idx1=f+1, idx2=f+2) so the
  // distance-2 row prefetch uses an index loaded a full iteration earlier:
  // no dependent wait inside the loop.
  {
    int idx0 = cat_indices[idxBase + 0];
    int idx1 = cat_indices[idxBase + 1];
    int idx2 = cat_indices[idxBase + 2];
    for (int f = 0; f < F_CAT; ++f) {
      // prefetch row f+2 into GL2 (covers HBM compulsory misses)
      const float* pp = embed_table + (long long)idx2 * D_DIM;
      __builtin_prefetch(pp, 0, 3);
      __builtin_prefetch(pp + 32, 0, 3);

      const float* rp = embed_table + (long long)idx0 * D_DIM + 2 * h;
      v2f a[16];
#pragma unroll
      for (int s = 0; s < 16; ++s) a[s] = *(const v2f*)(rp + 4 * s);

      // rotate index pipeline; idx2's new value is consumed next iteration
      idx0 = idx1;
      idx1 = idx2;
      if (f + 3 < F_CAT) idx2 = cat_indices[idxBase + f + 3];

#pragma unroll
      for (int s = 0; s < 16; ++s) {
        summed[s] += a[s];
        sumsq[s]  += a[s] * a[s];
      }
    }
  }

  // ---- S = summed @ ga_W  (row-sum of support over all 51 fields) ----
  v8f S[4];
#pragma unroll
  for (int t = 0; t < 4; ++t) {
    v8f c = {};
#pragma unroll
    for (int s = 0; s < 16; ++s) c = wmma4(summed[s], bGA[t][s], c);
    S[t] = c;
  }

  float gab[4];
#pragma unroll
  for (int t = 0; t < 4; ++t) gab[t] = ga_b[16 * t + m];

  // ---- pass 2: per-field support, relu-accumulate (L2-hot re-gather,
  //      latency hidden behind 64 WMMAs per field) ----
  v8f gacc[4];
#pragma unroll
  for (int t = 0; t < 4; ++t) gacc[t] = (v8f){};

  for (int f = 0; f <= F_CAT; ++f) {   // 50 shallow + 1 numeric field
    v2f a[16];
    if (f < F_CAT) {
      const int idx = cat_indices[idxBase + f];
      const float* rp = embed_table + (long long)idx * D_DIM + 2 * h;
#pragma unroll
      for (int s = 0; s < 16; ++s) a[s] = *(const v2f*)(rp + 4 * s);
    } else {
#pragma unroll
      for (int s = 0; s < 16; ++s) {
        const int d0 = 4 * s + 2 * h;
        v2f w  = *(const v2f*)(num_W + d0);
        v2f bb = *(const v2f*)(num_b + d0);
        a[s] = nf * w + bb;
      }
    }
#pragma unroll
    for (int t = 0; t < 4; ++t) {
      v8f c = {};
#pragma unroll
      for (int s = 0; s < 16; ++s) c = wmma4(a[s], bGA[t][s], c);
#pragma unroll
      for (int j = 0; j < 8; ++j) {
        const float gv = (S[t][j] + PROBE_C * c[j]) * INV_DEN + gab[t];
        gacc[t][j] += fmaxf(gv, 0.0f);
      }
    }
  }
#pragma unroll
  for (int t = 0; t < 4; ++t)
#pragma unroll
    for (int j = 0; j < 8; ++j) gacc[t][j] *= INV_NF;   // mean over 51 fields

  // ---- global MLP: two 64x64 dense layers ----
  v2f afr[16];
  c_store_lds(xb, m, h, gacc);        // C-layout -> LDS (in-order per wave)
  a_load_lds(xb, m, h, afr);
  v8f g1[4];
  dense64(afr, gW, gb, m, h, /*relu=*/true, g1);
  c_store_lds(xb, m, h, g1);
  a_load_lds(xb, m, h, afr);
  v8f g2[4];
  dense64(afr, gW + 64 * 64, gb + 64, m, h, /*relu=*/false, g2);

  // ---- local FM branch: 0.5*(summed^2 - sumsq), already in A-layout ----
  v2f al[16];
#pragma unroll
  for (int s = 0; s < 16; ++s)
    al[s] = 0.5f * (summed[s] * summed[s] - sumsq[s]);
  v8f l1[4];
  dense64(al, lW, lb, m, h, /*relu=*/true, l1);
  c_store_lds(xb, m, h, l1);
  a_load_lds(xb, m, h, al);
  v8f l2[4];
  dense64(al, lW + 64 * 64, lb + 64, m, h, /*relu=*/false, l2);

  // ---- out = 0.5*g + 0.5*l, stored from C-layout (16-lane b32 segments) ----
#pragma unroll
  for (int t = 0; t < 4; ++t)
#pragma unroll
    for (int r = 0; r < 8; ++r) {
      const int orow = row0 + r + 8 * h;
      out[(long long)orow * D_DIM + 16 * t + m] = 0.5f * (g2[t][r] + l2[t][r]);
    }
}

extern "C" void kernel_launch(void* const* d_in, const int* in_sizes, int n_in,
                              void* d_out, int out_size, void* d_ws, size_t ws_size,
                              hipStream_t stream) {
  (void)in_sizes; (void)n_in; (void)out_size; (void)d_ws; (void)ws_size;
  const int*   cat   = (const int*)  d_in[0];
  const float* nfeat = (const float*)d_in[1];
  const float* table = (const float*)d_in[2];
  const float* numW  = (const float*)d_in[3];
  const float* numb  = (const float*)d_in[4];
  const float* gaW   = (const float*)d_in[5];
  const float* gab   = (const float*)d_in[6];
  const float* gWp   = (const float*)d_in[7];
  const float* gbp   = (const float*)d_in[8];
  const float* lWp   = (const float*)d_in[9];
  const float* lbp   = (const float*)d_in[10];
  float* outp = (float*)d_out;

  dim3 grid(B_TOTAL / (WAVES_PER_WG * 16));   // 512
  dim3 block(WAVES_PER_WG * 32);              // 128 threads = 4 waves
  catembed_fused<<<grid, block, 0, stream>>>(cat, nfeat, table, numW, numb,
                                             gaW, gab, gWp, gbp, lWp, lbp, outp);
}